// NEQUIP_17540646437761
// MI455X (gfx1250) — compile-verified
//
#include <hip/hip_runtime.h>
#include <math.h>

// ---------------------------------------------------------------------------
// NequIP-style equivariant GNN forward for MI455X (gfx1250).
//  - Dense linear algebra via V_WMMA_F32_16X16X4_F32 (f32, matches reference),
//    dual accumulator chains to cover XDL latency.
//  - Fused edge kernel: per-path radial weights (WMMA on LDS-staged h2) +
//    CG tensor product + atomic scatter-add.
//  - CG tables computed on-device once per launch; staged into LDS by the
//    Tensor Data Mover (tensor_load_to_lds, TENSORcnt) in the message kernel.
// Feature layout: F[node][mslot(16)][chan(64)]; slots l0->0, l1->1..3,
// l2->4..8, l3->9..15.  All per-buffer offsets fit in 32 bits.
// ---------------------------------------------------------------------------

typedef __attribute__((ext_vector_type(2))) float v2f;
typedef __attribute__((ext_vector_type(8))) float v8f;
typedef __attribute__((ext_vector_type(4))) unsigned int u32x4;
typedef __attribute__((ext_vector_type(8))) unsigned int u32x8;

#define NPATHS_ALL 23
#define TEDGE 16
#define PCHUNK 8
#define CG_TOTAL 1875

// Path tables (l1,l2,l3); layer 0 uses the first 4 (l1==0) entries.
static constexpr int cL1[NPATHS_ALL] = {0,0,0,0, 1,1,1,1,1,1, 2,2,2,2,2,2,2, 3,3,3,3,3,3};
static constexpr int cL2[NPATHS_ALL] = {0,1,2,3, 0,1,1,2,2,3, 0,1,1,2,2,3,3, 0,1,2,2,3,3};
static constexpr int cL3[NPATHS_ALL] = {0,1,2,3, 1,0,2,1,3,2, 2,1,3,0,2,1,3, 3,2,1,3,0,2};
static constexpr int cCGOFF[NPATHS_ALL] = {0,1,10,35, 84,93,102,147,192,297,
                                           402,427,472,577,602,727,832,
                                           1077,1126,1231,1336,1581,1630};
static constexpr int cSLOT[4] = {0,1,4,9};

// Runtime-indexed copies for the CG init kernel.
__constant__ int dL1[NPATHS_ALL] = {0,0,0,0, 1,1,1,1,1,1, 2,2,2,2,2,2,2, 3,3,3,3,3,3};
__constant__ int dL2[NPATHS_ALL] = {0,1,2,3, 0,1,1,2,2,3, 0,1,1,2,2,3,3, 0,1,2,2,3,3};
__constant__ int dL3[NPATHS_ALL] = {0,1,2,3, 1,0,2,1,3,2, 2,1,3,0,2,1,3, 3,2,1,3,0,2};
__constant__ int dCGOFF[NPATHS_ALL] = {0,1,10,35, 84,93,102,147,192,297,
                                       402,427,472,577,602,727,832,
                                       1077,1126,1231,1336,1581,1630};

__device__ __forceinline__ float dev_silu(float x) { return x / (1.f + __expf(-x)); }
__device__ __forceinline__ float dev_sigmoid(float x) { return 1.f / (1.f + __expf(-x)); }

// ---------------------------------------------------------------------------
// CG table construction (device, double precision, one thread per path)
// ---------------------------------------------------------------------------
__device__ double dfac(int n) { double f = 1.0; for (int i = 2; i <= n; ++i) f *= (double)i; return f; }

__device__ void buildU(int l, double* Ure, double* Uim) {
  int d = 2 * l + 1;
  for (int i = 0; i < d * d; i++) { Ure[i] = 0.0; Uim[i] = 0.0; }
  Ure[l * d + l] = 1.0;
  const double is2 = 0.70710678118654752440;
  for (int m = 1; m <= l; m++) {
    double sgn = (m & 1) ? -1.0 : 1.0;
    Ure[(l + m) * d + (l + m)] = sgn * is2;
    Ure[(l + m) * d + (l - m)] = is2;
    Uim[(l - m) * d + (l - m)] = is2;
    Uim[(l - m) * d + (l + m)] = -sgn * is2;
  }
}

__global__ void k_cg_init(float* cgRe, float* cgIm) {
  int p = threadIdx.x;
  if (p >= NPATHS_ALL) return;
  int j1 = dL1[p], j2 = dL2[p], j3 = dL3[p];
  int d1 = 2 * j1 + 1, d2 = 2 * j2 + 1, d3 = 2 * j3 + 1;
  double T[343];
  for (int i = 0; i < d1 * d2 * d3; i++) T[i] = 0.0;
  double pref0 = sqrt((double)(2 * j3 + 1) * dfac(j1 + j2 - j3) * dfac(j1 - j2 + j3) *
                      dfac(-j1 + j2 + j3) / dfac(j1 + j2 + j3 + 1));
  for (int m1 = -j1; m1 <= j1; m1++)
    for (int m2 = -j2; m2 <= j2; m2++) {
      int m3 = m1 + m2;
      if (m3 < -j3 || m3 > j3) continue;
      double pref = pref0 * sqrt(dfac(j1 + m1) * dfac(j1 - m1) * dfac(j2 + m2) *
                                 dfac(j2 - m2) * dfac(j3 + m3) * dfac(j3 - m3));
      double s = 0.0;
      for (int k = 0; k <= j1 + j2 - j3; k++) {
        int a0 = k, a1 = j1 + j2 - j3 - k, a2 = j1 - m1 - k, a3 = j2 + m2 - k;
        int a4 = j3 - j2 + m1 + k, a5 = j3 - j1 - m2 + k;
        if (a0 < 0 || a1 < 0 || a2 < 0 || a3 < 0 || a4 < 0 || a5 < 0) continue;
        double term = 1.0 / (dfac(a0) * dfac(a1) * dfac(a2) * dfac(a3) * dfac(a4) * dfac(a5));
        s += (k & 1) ? -term : term;
      }
      T[((m1 + j1) * d2 + (m2 + j2)) * d3 + (m3 + j3)] = pref * s;
    }
  double U1re[49], U1im[49], U2re[49], U2im[49], U3re[49], U3im[49];
  buildU(j1, U1re, U1im); buildU(j2, U2re, U2im); buildU(j3, U3re, U3im);
  double mre = 0.0, mim = 0.0;
  int off = dCGOFF[p];
  for (int a = 0; a < d1; a++)
    for (int b = 0; b < d2; b++)
      for (int c = 0; c < d3; c++) {
        double re = 0.0, im = 0.0;
        for (int m = 0; m < d1; m++) {
          double ar = U1re[a * d1 + m], ai = U1im[a * d1 + m];
          if (ar == 0.0 && ai == 0.0) continue;
          for (int n = 0; n < d2; n++) {
            double br = U2re[b * d2 + n], bi = U2im[b * d2 + n];
            if (br == 0.0 && bi == 0.0) continue;
            double abr = ar * br - ai * bi, abi = ar * bi + ai * br;
            for (int o = 0; o < d3; o++) {
              double t = T[(m * d2 + n) * d3 + o];
              if (t == 0.0) continue;
              double cr = U3re[c * d3 + o], ci = -U3im[c * d3 + o]; // conj
              re += (abr * cr - abi * ci) * t;
              im += (abr * ci + abi * cr) * t;
            }
          }
        }
        cgRe[off + (a * d2 + b) * d3 + c] = (float)re;
        cgIm[off + (a * d2 + b) * d3 + c] = (float)im;
        mre = fmax(mre, fabs(re)); mim = fmax(mim, fabs(im));
      }
  if (mim > mre)
    for (int i = 0; i < d1 * d2 * d3; i++) cgRe[off + i] = cgIm[off + i];
}

// ---------------------------------------------------------------------------
// Elementwise helpers
// ---------------------------------------------------------------------------
__global__ void k_zero(float* p, int n) {
  int i = blockIdx.x * 256 + threadIdx.x;
  if (i < n) p[i] = 0.f;
}

__global__ void k_embed_gather(const float* __restrict__ embed, const int* __restrict__ species,
                               float* __restrict__ F0, int N) {
  int i = blockIdx.x * 256 + threadIdx.x;
  if (i >= N * 32) return;
  int n = i >> 5, c = i & 31;
  F0[i] = embed[species[n] * 32 + c];
}

// Per-edge geometry: spherical harmonics (16) + radial basis (8).
__global__ __launch_bounds__(128) void k_edge_geom(const float* __restrict__ pos,
                                                   const int* __restrict__ snd,
                                                   const int* __restrict__ rcv,
                                                   float* __restrict__ shG,
                                                   float* __restrict__ rbG, int E) {
  int e = blockIdx.x * 128 + threadIdx.x;
  if (e >= E) return;
  int s = snd[e], r = rcv[e];
  float vx = pos[r * 3 + 0] - pos[s * 3 + 0];
  float vy = pos[r * 3 + 1] - pos[s * 3 + 1];
  float vz = pos[r * 3 + 2] - pos[s * 3 + 2];
  float rn = sqrtf(vx * vx + vy * vy + vz * vz);
  float ir = (rn > 0.f) ? 1.f / rn : 1.f;
  float x = vx * ir, y = vy * ir, z = vz * ir;
  float* sh = shG + e * 16;
  const float s3 = 1.7320508075688772f;
  const float s15 = 3.872983346207417f, s5h = 1.118033988749895f, s15h = 1.9364916731037085f;
  const float c70 = 2.091650066335189f, c105 = 10.246950765959598f;
  const float c42 = 1.6201851746019651f, c7h = 1.3228756555322954f, c105h = 5.123475382979799f;
  sh[0] = 1.f;
  sh[1] = s3 * y; sh[2] = s3 * z; sh[3] = s3 * x;
  sh[4] = s15 * x * y; sh[5] = s15 * y * z; sh[6] = s5h * (3.f * z * z - 1.f);
  sh[7] = s15 * x * z; sh[8] = s15h * (x * x - y * y);
  sh[9]  = c70 * y * (3.f * x * x - y * y);
  sh[10] = c105 * x * y * z;
  sh[11] = c42 * y * (5.f * z * z - 1.f);
  sh[12] = c7h * z * (5.f * z * z - 3.f);
  sh[13] = c42 * x * (5.f * z * z - 1.f);
  sh[14] = c105h * z * (x * x - y * y);
  sh[15] = c70 * x * (x * x - 3.f * y * y);
  const float RM = 2.0f, PI = 3.14159265358979323846f;
  float rs = (rn > 0.f) ? rn : 1.f;
  float xr = rn / RM;
  float x2 = xr * xr, x6 = x2 * x2 * x2, x7 = x6 * xr, x8 = x7 * xr;
  float env = 1.f - 28.f * x6 + 48.f * x7 - 21.f * x8;
  if (xr >= 1.f) env = 0.f;
  float* rb = rbG + e * 8;
#pragma unroll
  for (int n = 1; n <= 8; n++)
    rb[n - 1] = sinf((float)n * PI * rs / RM) / rs * env;
}

// ---------------------------------------------------------------------------
// Generic f32 WMMA GEMM: C[r,n] = act(sum_k A[r,k]*B[k,n] + bias[n])
// Row r -> memory via (r/DEG)*aOuter + (r%DEG)*aInner (K contiguous).
// DEG is a template constant; 32-bit offsets; pointer-bumped B streams;
// two independent WMMA accumulator chains (K is a multiple of 8 here).
// Block = 256 threads = 8 waves = 2x4 grid of 16x16 tiles. M multiple of 16.
// ---------------------------------------------------------------------------
template <int DEG>
__global__ __launch_bounds__(256) void k_gemm(const float* __restrict__ A, int aOuter, int aInner,
                                              const float* __restrict__ B, int ldb,
                                              const float* __restrict__ bias, int act,
                                              float* __restrict__ C, int cOuter, int cInner,
                                              int M, int Nc, int K) {
  int tid = threadIdx.x;
  int wv = tid >> 5, lane = tid & 31;
  int half = lane >> 4, l16 = lane & 15;
  int mT = wv >> 2, nT = wv & 3;
  int Mbase = blockIdx.x * 32 + mT * 16;
  int Nbase = blockIdx.y * 64 + nT * 16;
  if (Mbase >= M || Nbase >= Nc) return;
  int rowA = Mbase + l16;
  const float* Ar = A + (rowA / DEG) * aOuter + (rowA % DEG) * aInner;
  // Four zero-offset B streams: {chain0,chain1} x {k0, k0+1}, bumped by 8*ldb.
  const float* B00 = B + Nbase + l16 + (half * 2) * ldb;
  const float* B01 = B00 + ldb;
  const float* B10 = B00 + 4 * ldb;
  const float* B11 = B10 + ldb;
  __builtin_prefetch(Ar, 0, 1);
  __builtin_prefetch(B00, 0, 1);
  v8f acc0 = {}, acc1 = {};
  for (int k = 0; k < K; k += 8) {
    int k0 = k + half * 2;
    v2f a0, b0, a1, b1;
    a0[0] = Ar[k0]; a0[1] = Ar[k0 + 1];
    b0[0] = *B00;   b0[1] = *B01;
    acc0 = __builtin_amdgcn_wmma_f32_16x16x4_f32(false, a0, false, b0, (short)0, acc0, false, false);
    a1[0] = Ar[k0 + 4]; a1[1] = Ar[k0 + 5];
    b1[0] = *B10;       b1[1] = *B11;
    acc1 = __builtin_amdgcn_wmma_f32_16x16x4_f32(false, a1, false, b1, (short)0, acc1, false, false);
    B00 += 8 * ldb; B01 += 8 * ldb; B10 += 8 * ldb; B11 += 8 * ldb;
  }
  int colC = Nbase + l16;
  float bv = bias ? bias[colC] : 0.f;
#pragma unroll
  for (int v = 0; v < 8; v++) {
    int rowC = Mbase + half * 8 + v;
    float xv = acc0[v] + acc1[v] + bv;
    if (act == 1) xv = dev_silu(xv);
    C[(rowC / DEG) * cOuter + (rowC % DEG) * cInner + colC] = xv;
  }
}

// ---------------------------------------------------------------------------
// Fused message kernel: per 16-edge tile, CG table DMA'd to LDS by the TDM,
// path weights w = h2 @ Wr3 slice via WMMA into LDS (8-path chunks), then CG
// tensor product + atomic scatter-add.
// ---------------------------------------------------------------------------
template <int LAYER>
__global__ __launch_bounds__(256) void k_message(const float* __restrict__ X,
                                                 const float* __restrict__ H2,
                                                 const float* __restrict__ Wr3,
                                                 const float* __restrict__ shG,
                                                 const int* __restrict__ senders,
                                                 const int* __restrict__ receivers,
                                                 const float* __restrict__ CGbuf,
                                                 float* __restrict__ AGG, int E) {
  constexpr int NP = (LAYER == 0) ? 4 : 23;
  constexpr int NSLOT = (LAYER == 0) ? 1 : 16;
  constexpr int LDW = NP * 64;
  __shared__ float sCG[CG_TOTAL];
  __shared__ float sSH[TEDGE * 16];
  __shared__ float sH[TEDGE * 64];
  __shared__ float sW[PCHUNK * TEDGE * 64];
  int tid = threadIdx.x;
  int eBase = blockIdx.x * TEDGE;
  int wv = tid >> 5, lane = tid & 31;

  // --- Tensor Data Mover: DMA the 7.5KB CG table global->LDS (wave 0 only),
  //     overlapped with the vector-loaded sh/h2 tiles below. ---
  if (wv == 0) {
    unsigned long long g = (unsigned long long)CGbuf;
    unsigned ldsAddr = (unsigned)(unsigned long long)(&sCG[0]);
    u32x4 g0;
    g0[0] = 1u;                                   // count=1, user descriptor
    g0[1] = ldsAddr;                              // lds_addr
    g0[2] = (unsigned)g;                          // global_addr[31:0]
    g0[3] = (unsigned)((g >> 32) & 0x01FFFFFFull) // global_addr[56:32]
            | (2u << 30);                         // type=2 ("image")
    u32x8 g1;
    g1[0] = 2u << 16;                             // data_size=2 (4 bytes)
    g1[1] = ((unsigned)CG_TOTAL & 0xFFFFu) << 16; // tensor_dim0[15:0]
    g1[2] = ((unsigned)CG_TOTAL >> 16)            // tensor_dim0[31:16]
            | (1u << 16);                         // tensor_dim1[15:0] = 1
    g1[3] = ((unsigned)CG_TOTAL) << 16;           // tile_dim0 = 1875
    g1[4] = 0u;                                   // tile_dim1=0, tile_dim2=0
    g1[5] = (unsigned)CG_TOTAL;                   // tensor_dim0_stride[31:0]
    g1[6] = 0u;
    g1[7] = 0u;
    asm volatile("tensor_load_to_lds %0, %1" :: "s"(g0), "s"(g1) : "memory");
  }
  for (int i = tid; i < TEDGE * 16; i += 256) {
    int e = eBase + (i >> 4);
    sSH[i] = (e < E) ? shG[e * 16 + (i & 15)] : 0.f;
  }
  for (int i = tid; i < TEDGE * 64; i += 256) {
    int e = eBase + (i >> 6);
    sH[i] = (e < E) ? H2[e * 64 + (i & 63)] : 0.f;
  }
  if (wv == 0) __builtin_amdgcn_s_wait_tensorcnt(0);
  __syncthreads();

  int eLoc = tid >> 4, c16 = tid & 15;
  bool ve = (eBase + eLoc) < E;
  int snd = ve ? senders[eBase + eLoc] : 0;
  int rcv = ve ? receivers[eBase + eLoc] : 0;
  int half = lane >> 4, l16 = lane & 15;

#pragma unroll
  for (int p0 = 0; p0 < NP; p0 += PCHUNK) {
    const int pc = (NP - p0 < PCHUNK) ? (NP - p0) : PCHUNK;
    __syncthreads();
    // --- path weight GEMM for this chunk: [16e x 64k] @ [64k x pc*64] ---
    for (int t = wv; t < pc * 4; t += 8) {
      int pLoc = t >> 2, nt = t & 3;
      int pi = p0 + pLoc;
      const float* Bp = Wr3 + pi * 64 + nt * 16 + l16 + (half * 2) * LDW;
      __builtin_prefetch(Bp, 0, 1);
      v8f acc0 = {}, acc1 = {};
#pragma unroll
      for (int k = 0; k < 64; k += 8) {
        int k0 = k + half * 2;
        v2f a0, b0, a1, b1;
        a0[0] = sH[l16 * 64 + k0]; a0[1] = sH[l16 * 64 + k0 + 1];
        b0[0] = Bp[k * LDW]; b0[1] = Bp[k * LDW + LDW];
        acc0 = __builtin_amdgcn_wmma_f32_16x16x4_f32(false, a0, false, b0, (short)0, acc0, false, false);
        a1[0] = sH[l16 * 64 + k0 + 4]; a1[1] = sH[l16 * 64 + k0 + 5];
        b1[0] = Bp[(k + 4) * LDW]; b1[1] = Bp[(k + 4) * LDW + LDW];
        acc1 = __builtin_amdgcn_wmma_f32_16x16x4_f32(false, a1, false, b1, (short)0, acc1, false, false);
      }
#pragma unroll
      for (int v = 0; v < 8; v++) {
        int row = half * 8 + v;
        sW[(pLoc * TEDGE + row) * 64 + nt * 16 + l16] = acc0[v] + acc1[v];
      }
    }
    __syncthreads();
    // --- tensor product + scatter, 4 channel passes ---
    for (int cp = 0; cp < 4; cp++) {
      int c = cp * 16 + c16;
      float xs[16];
#pragma unroll
      for (int m = 0; m < 16; m++)
        xs[m] = (m < NSLOT) ? X[(snd * 16 + m) * 64 + c] : 0.f;
      float acm[16];
#pragma unroll
      for (int m = 0; m < 16; m++) acm[m] = 0.f;
#pragma unroll
      for (int pp = 0; pp < PCHUNK; pp++) {
        if (pp >= pc) break;
        const int pi = p0 + pp;
        const int l1 = cL1[pi], l2 = cL2[pi], l3 = cL3[pi];
        const int d1 = 2 * l1 + 1, d2 = 2 * l2 + 1, d3 = 2 * l3 + 1;
        const float* cg = &sCG[cCGOFF[pi]];
        float wgt = sW[(pp * TEDGE + eLoc) * 64 + c];
        const float* she = &sSH[eLoc * 16 + cSLOT[l2]];
#pragma unroll
        for (int i2 = 0; i2 < d1; i2++) {
          float xw = xs[cSLOT[l1] + i2] * wgt;
#pragma unroll
          for (int j = 0; j < d2; j++) {
            float xsh = xw * she[j];
#pragma unroll
            for (int k = 0; k < d3; k++)
              acm[cSLOT[l3] + k] += xsh * cg[(i2 * d2 + j) * d3 + k];
          }
        }
      }
      // Every chunk's path set touches all 16 m-slots -> unconditional scatter.
      if (ve) {
#pragma unroll
        for (int m = 0; m < 16; m++)
          atomicAdd(&AGG[(rcv * 16 + m) * 64 + c], acm[m]);
      }
    }
  }
}

// ---------------------------------------------------------------------------
// Node update tail: gates, species-dependent skip, activations.
// One block (64 threads = channels) per node.
// ---------------------------------------------------------------------------
__global__ __launch_bounds__(64) void k_finalize(const float* __restrict__ S,
                                                 const float* __restrict__ Y,
                                                 const float* __restrict__ Fprev,
                                                 const float* __restrict__ sk0w,
                                                 const float* __restrict__ sk1w,
                                                 const float* __restrict__ sk2w,
                                                 const float* __restrict__ sk3w,
                                                 const int* __restrict__ species,
                                                 float* __restrict__ Fnext,
                                                 int cin, int fullSkip) {
  __shared__ float sF[1024];
  int n = blockIdx.x, d = threadIdx.x;
  int sp = species[n];
  int nload = fullSkip ? 1024 : cin;
  const float* Fr = fullSkip ? (Fprev + n * 1024) : (Fprev + n * cin);
  for (int i = d; i < nload; i += 64) sF[i] = Fr[i];
  __syncthreads();
  float f0 = S[n * 256 + d * 4];
  {
    const float* w0 = sk0w + sp * cin * 64;
    float sk = 0.f;
    for (int c = 0; c < cin; c++) sk += sF[c] * w0[c * 64 + d];
    f0 += sk;
  }
  Fnext[n * 1024 + d] = dev_silu(f0);
#pragma unroll
  for (int l = 1; l <= 3; l++) {
    float g = dev_sigmoid(S[n * 256 + d * 4 + l]);
    const int off = (l == 1) ? 1 : (l == 2) ? 4 : 9;
    const int deg = 2 * l + 1;
    const float* wl = (l == 1) ? sk1w : (l == 2) ? sk2w : sk3w;
    const float* wls = fullSkip ? (wl + sp * 64 * 64) : nullptr;
    for (int m = 0; m < deg; m++) {
      float v = Y[(n * 16 + off + m) * 64 + d];
      if (fullSkip) {
        float sk = 0.f;
        const float* fr = &sF[(off + m) * 64];
        for (int c = 0; c < 64; c++) sk += fr[c] * wls[c * 64 + d];
        v += sk;
      }
      Fnext[(n * 16 + off + m) * 64 + d] = v * g;
    }
  }
}

// Transpose-concat to the reference output layout: [f0(64) | l1(64x3) | l2(64x5) | l3(64x7)]
__global__ __launch_bounds__(256) void k_output(const float* __restrict__ F, float* __restrict__ out, int N) {
  int i = blockIdx.x * 256 + threadIdx.x;
  int total = N * 1024;
  if (i >= total) return;
  int n = i / 1024;
  int r = i % 1024;
  int c, m, off;
  if (r < 64)       { c = r; m = 0; off = 0; }
  else if (r < 256) { int t = r - 64;  c = t / 3; m = t % 3; off = 1; }
  else if (r < 576) { int t = r - 256; c = t / 5; m = t % 5; off = 4; }
  else              { int t = r - 576; c = t / 7; m = t % 7; off = 9; }
  out[i] = F[(n * 16 + off + m) * 64 + c];
}

// ---------------------------------------------------------------------------
// Host side
// ---------------------------------------------------------------------------
static void gemm(hipStream_t s, int deg, const float* A, int aOuter, int aInner,
                 const float* B, int ldb, const float* bias, int act,
                 float* C, int cOuter, int cInner, int M, int Nc, int K) {
  dim3 grid((M + 31) / 32, Nc / 64);
  switch (deg) {
    case 1: k_gemm<1><<<grid, 256, 0, s>>>(A, aOuter, aInner, B, ldb, bias, act, C, cOuter, cInner, M, Nc, K); break;
    case 3: k_gemm<3><<<grid, 256, 0, s>>>(A, aOuter, aInner, B, ldb, bias, act, C, cOuter, cInner, M, Nc, K); break;
    case 5: k_gemm<5><<<grid, 256, 0, s>>>(A, aOuter, aInner, B, ldb, bias, act, C, cOuter, cInner, M, Nc, K); break;
    default: k_gemm<7><<<grid, 256, 0, s>>>(A, aOuter, aInner, B, ldb, bias, act, C, cOuter, cInner, M, Nc, K); break;
  }
}

extern "C" void kernel_launch(void* const* d_in, const int* in_sizes, int n_in,
                              void* d_out, int out_size, void* d_ws, size_t ws_size,
                              hipStream_t stream) {
  const float* pos = (const float*)d_in[0];
  const int* species = (const int*)d_in[1];
  const int* senders = (const int*)d_in[2];
  const int* receivers = (const int*)d_in[3];
  const float* embed = (const float*)d_in[4];
  const int N = in_sizes[1];
  const int E = in_sizes[2];

  // Params flattened in jax pytree order (dict keys sorted: W* < b* < lin* < skip).
  struct LayerP {
    const float *Wr1, *Wr2, *Wr3, *br1, *br2;
    const float* lin1[4];
    const float* lin2[4];
    const float* skip[4];
    int npaths, cin, nls;
  } L[3];
  int idx = 5;
  for (int li = 0; li < 3; li++) {
    LayerP& lp = L[li];
    lp.nls = (li == 0) ? 1 : 4;
    lp.cin = (li == 0) ? 32 : 64;
    lp.npaths = (li == 0) ? 4 : 23;
    for (int l = 0; l < 4; l++) { lp.lin1[l] = nullptr; lp.skip[l] = nullptr; }
    lp.Wr1 = (const float*)d_in[idx++];
    lp.Wr2 = (const float*)d_in[idx++];
    lp.Wr3 = (const float*)d_in[idx++];
    lp.br1 = (const float*)d_in[idx++];
    lp.br2 = (const float*)d_in[idx++];
    for (int l = 0; l < lp.nls; l++) lp.lin1[l] = (const float*)d_in[idx++];
    for (int l = 0; l < 4; l++) lp.lin2[l] = (const float*)d_in[idx++];
    for (int l = 0; l < lp.nls; l++) lp.skip[l] = (const float*)d_in[idx++];
  }

  // Workspace carve (floats).
  float* W = (float*)d_ws;
  size_t o = 0;
  float* SH = W + o; o += (size_t)E * 16;
  float* RB = W + o; o += (size_t)E * 8;
  float* H1 = W + o; o += (size_t)E * 64;
  float* H2 = W + o; o += (size_t)E * 64;
  float* F0 = W + o; o += (size_t)N * 32;
  float* FA = W + o; o += (size_t)N * 1024;
  float* FB = W + o; o += (size_t)N * 1024;
  float* Xb = W + o; o += (size_t)N * 1024;
  float* AG = W + o; o += (size_t)N * 1024;
  float* Yb = W + o; o += (size_t)N * 1024;
  float* Sb = W + o; o += (size_t)N * 256;
  float* CGre = W + o; o += CG_TOTAL;
  float* CGim = W + o; o += CG_TOTAL;
  (void)ws_size; (void)n_in; (void)out_size;

  k_cg_init<<<1, 32, 0, stream>>>(CGre, CGim);
  k_embed_gather<<<(N * 32 + 255) / 256, 256, 0, stream>>>(embed, species, F0, N);
  k_edge_geom<<<(E + 127) / 128, 128, 0, stream>>>(pos, senders, receivers, SH, RB, E);

  const int OFFh[4] = {0, 1, 4, 9};
  const float* Fprev = F0;
  float* Fnext = FA;
  for (int li = 0; li < 3; li++) {
    LayerP& lp = L[li];
    // lin1: x[l] = feats[l] @ W  (rows = (node, m), K = channels)
    if (li == 0) {
      gemm(stream, 1, Fprev, 32, 0, lp.lin1[0], 64, nullptr, 0, Xb, 1024, 0, N, 64, 32);
    } else {
      for (int l = 0; l < 4; l++) {
        int deg = 2 * l + 1;
        gemm(stream, deg, Fprev + OFFh[l] * 64, 1024, 64, lp.lin1[l], 64, nullptr, 0,
             Xb + OFFh[l] * 64, 1024, 64, N * deg, 64, 64);
      }
    }
    // radial MLP h2 = silu(silu(rb@Wr1+b1)@Wr2+b2)
    gemm(stream, 1, RB, 8, 0, lp.Wr1, 64, lp.br1, 1, H1, 64, 0, E, 64, 8);
    gemm(stream, 1, H1, 64, 0, lp.Wr2, 64, lp.br2, 1, H2, 64, 0, E, 64, 64);
    // messages
    k_zero<<<(N * 1024 + 255) / 256, 256, 0, stream>>>(AG, N * 1024);
    if (li == 0)
      k_message<0><<<(E + TEDGE - 1) / TEDGE, 256, 0, stream>>>(Xb, H2, lp.Wr3, SH, senders,
                                                                receivers, CGre, AG, E);
    else
      k_message<1><<<(E + TEDGE - 1) / TEDGE, 256, 0, stream>>>(Xb, H2, lp.Wr3, SH, senders,
                                                                receivers, CGre, AG, E);
    // node update GEMMs
    gemm(stream, 1, AG, 1024, 0, lp.lin2[0], 256, nullptr, 0, Sb, 256, 0, N, 256, 64);
    for (int l = 1; l <= 3; l++) {
      int deg = 2 * l + 1;
      gemm(stream, deg, AG + OFFh[l] * 64, 1024, 64, lp.lin2[l], 64, nullptr, 0,
           Yb + OFFh[l] * 64, 1024, 64, N * deg, 64, 64);
    }
    k_finalize<<<N, 64, 0, stream>>>(Sb, Yb, Fprev, lp.skip[0], lp.skip[1], lp.skip[2], lp.skip[3],
                                     species, Fnext, lp.cin, (li == 0) ? 0 : 1);
    Fprev = Fnext;
    Fnext = (Fnext == FA) ? FB : FA;
  }
  k_output<<<(N * 1024 + 255) / 256, 256, 0, stream>>>(Fprev, (float*)d_out, N);
}